// LF4DGrid_70471823393087
// MI455X (gfx1250) — compile-verified
//
#include <hip/hip_runtime.h>

// LF4DGrid: 4D quadrilinear grid interpolation, C=16 channels, grid 48^4, N=524288 rays.
// Strategy (MI455X, compile-only reasoning):
//  1) One transpose pass grid(C,P) -> grid_t(P,C) in d_ws: turns the 16-channel
//     corner gather into a single contiguous 64B read; two last-dim-adjacent
//     corners share one 128B line. ~0.5GB gather + 0.68GB transpose ~= 50us at
//     23.3 TB/s vs multi-GB traffic in the channel-major layout.
//  2) Per 16-ray tile, the weighted reduction out[n,c] = sum_k w[n,k]*v[n,k,c]
//     is computed exactly as a block-diagonal f32 GEMM with 64x
//     V_WMMA_F32_16X16X4_F32 (compute is ~100x under the memory roof, so the
//     sparse-A utilization is free and accumulation stays full f32).

typedef __attribute__((ext_vector_type(2))) float v2f;
typedef __attribute__((ext_vector_type(8))) float v8f;

#define GD  48
#define ST0 110592   // 48^3 stride of dim0
#define ST1 2304     // 48^2
#define ST2 48
#define GP  5308416  // 48^4
#define NCH 16

constexpr int RSTRIDE     = 260;                  // padded dwords per ray row in LDS (16*16 + 4)
constexpr int WAVE_FLOATS = 16 * RSTRIDE + 256;   // value tile + 16x16 weight tile

// ---------------- transpose: grid(C,P) -> grid_t(P,C) ----------------
__global__ __launch_bounds__(256) void lf4d_transpose(const float* __restrict__ g,
                                                      float* __restrict__ gt) {
  size_t p = (size_t)blockIdx.x * 256 + threadIdx.x;
  float v[NCH];
#pragma unroll
  for (int c = 0; c < NCH; ++c) v[c] = g[(size_t)c * GP + p];  // coalesced per-channel streams
  float4* o = (float4*)(gt + (p << 4));                        // 64B contiguous per lane
#pragma unroll
  for (int j = 0; j < 4; ++j) {
    float4 q;
    q.x = v[4 * j + 0]; q.y = v[4 * j + 1]; q.z = v[4 * j + 2]; q.w = v[4 * j + 3];
    o[j] = q;
  }
}

__device__ __forceinline__ void dim_setup(float coord, float& w0, float& w1, int& bcl) {
  float f = floorf(coord);
  int   b = (int)f;
  float w = coord - f;
  bool ok0 = (b >= 0) && (b < GD);
  bool ok1 = ((b + 1) >= 0) && ((b + 1) < GD);
  w0  = ok0 ? (1.0f - w) : 0.0f;   // invalid corner -> weight 0 (matches reference mask)
  w1  = ok1 ? w : 0.0f;
  bcl = b < 0 ? 0 : (b > GD - 2 ? GD - 2 : b);   // clamp so all 16 addresses stay in range
}

// ---------------- main kernel: transposed-layout gather + WMMA reduce ----------------
__global__ __launch_bounds__(64) void lf4d_wmma(const float* __restrict__ ray,
                                                const float* __restrict__ gt,
                                                const float* __restrict__ rmin,
                                                const float* __restrict__ rmax,
                                                float* __restrict__ out) {
  __shared__ __align__(16) float smem[2 * WAVE_FLOATS];   // 2 waves/block, 35.3 KB
  const int tid  = threadIdx.x;
  const int lane = tid & 31;
  const int wv   = tid >> 5;
  float* hb   = smem + wv * WAVE_FLOATS;   // [16 rays][16 corners][16 ch], ray stride 260
  float* wlds = hb + 16 * RSTRIDE;         // [16 rays][16 corner weights]
  const int r  = lane & 15;                // ray-in-tile (gather) == matrix column (WMMA)
  const int hi = lane >> 4;                // half selector: channels 8*hi.. / K rows 2*hi..

  const float4 rm = *(const float4*)rmin;
  const float4 rx = *(const float4*)rmax;
  const float s0 = (float)(GD - 1) / (rx.x - rm.x);
  const float s1 = (float)(GD - 1) / (rx.y - rm.y);
  const float s2 = (float)(GD - 1) / (rx.z - rm.z);
  const float s3 = (float)(GD - 1) / (rx.w - rm.w);

  for (int half = 0; half < 2; ++half) {
    const int tileBase = blockIdx.x * 64 + wv * 32 + half * 16;

    // ---- gather phase: 2 lanes per ray, lane pair fetches one 64B corner chunk ----
    const float4 rv = ((const float4*)ray)[tileBase + r];
    float w0[4], w1[4]; int bc[4];
    dim_setup((rv.x - rm.x) * s0, w0[0], w1[0], bc[0]);
    dim_setup((rv.y - rm.y) * s1, w0[1], w1[1], bc[1]);
    dim_setup((rv.z - rm.z) * s2, w0[2], w1[2], bc[2]);
    dim_setup((rv.w - rm.w) * s3, w0[3], w1[3], bc[3]);
    const int bilin = bc[0] * ST0 + bc[1] * ST1 + bc[2] * ST2 + bc[3];

#pragma unroll
    for (int k = 0; k < 16; ++k) {
      const int lin = bilin + ((k & 1) ? ST0 : 0) + ((k & 2) ? ST1 : 0)
                            + ((k & 4) ? ST2 : 0) + ((k & 8) ? 1 : 0);
      const float4* sp = (const float4*)(gt + ((size_t)lin << 4)) + hi * 2;
      const float4 a = sp[0];
      const float4 b = sp[1];
      float* dp = hb + r * RSTRIDE + k * 16 + hi * 8;
      *(float4*)dp       = a;
      *((float4*)dp + 1) = b;
      if (hi == 0) {
        const float wk = ((k & 1) ? w1[0] : w0[0]) * ((k & 2) ? w1[1] : w0[1])
                       * ((k & 4) ? w1[2] : w0[2]) * ((k & 8) ? w1[3] : w0[3]);
        wlds[r * 16 + k] = wk;
      }
    }
    __syncthreads();

    // ---- WMMA phase: D(16x16) = block-diag A(16x256) x B(256x16), K in steps of 4 ----
    v8f acc = {0.f, 0.f, 0.f, 0.f, 0.f, 0.f, 0.f, 0.f};
    const int rowoff = hi * 2;   // A/B layout: lanes 16-31 hold K=2,3 of each step
    for (int n0 = 0; n0 < 16; ++n0) {
      const bool  hit = (r == n0);                       // A row n0 is the only nonzero row
      const float* bb = hb + n0 * RSTRIDE + rowoff * 16 + r;
      const float* wb = wlds + n0 * 16 + rowoff;
#pragma unroll
      for (int s = 0; s < 4; ++s) {
        const int kb = 4 * s;
        v2f A, B;
        A.x = hit ? wb[kb]     : 0.0f;   // A[m][rowoff]   = w[n0][kb+rowoff]   iff m==n0
        A.y = hit ? wb[kb + 1] : 0.0f;   // A[m][rowoff+1] = w[n0][kb+rowoff+1] iff m==n0
        B.x = bb[kb * 16];               // B[rowoff][c]   = vals[n0][kb+rowoff][c]
        B.y = bb[kb * 16 + 16];          // B[rowoff+1][c]
        acc = __builtin_amdgcn_wmma_f32_16x16x4_f32(false, A, false, B,
                                                    (short)0, acc, false, false);
      }
    }

    // D layout: VGPR j -> rows j (lanes 0-15) / j+8 (lanes 16-31), col = lane&15
#pragma unroll
    for (int j = 0; j < 8; ++j)
      out[(size_t)(tileBase + hi * 8 + j) * NCH + r] = acc[j];
    __syncthreads();
  }
}

// ---------------- fallback: direct channel-major gather (no workspace) ----------------
__global__ __launch_bounds__(256) void lf4d_direct(const float* __restrict__ ray,
                                                   const float* __restrict__ g,
                                                   const float* __restrict__ rmin,
                                                   const float* __restrict__ rmax,
                                                   float* __restrict__ out, int N) {
  int n = blockIdx.x * 256 + threadIdx.x;
  if (n >= N) return;
  const float4 rm = *(const float4*)rmin;
  const float4 rx = *(const float4*)rmax;
  const float4 rv = ((const float4*)ray)[n];
  float w0[4], w1[4]; int bc[4];
  dim_setup((rv.x - rm.x) * ((float)(GD - 1) / (rx.x - rm.x)), w0[0], w1[0], bc[0]);
  dim_setup((rv.y - rm.y) * ((float)(GD - 1) / (rx.y - rm.y)), w0[1], w1[1], bc[1]);
  dim_setup((rv.z - rm.z) * ((float)(GD - 1) / (rx.z - rm.z)), w0[2], w1[2], bc[2]);
  dim_setup((rv.w - rm.w) * ((float)(GD - 1) / (rx.w - rm.w)), w0[3], w1[3], bc[3]);
  const int bilin = bc[0] * ST0 + bc[1] * ST1 + bc[2] * ST2 + bc[3];

  float acc[NCH];
#pragma unroll
  for (int c = 0; c < NCH; ++c) acc[c] = 0.0f;
#pragma unroll
  for (int k = 0; k < 16; ++k) {
    const int lin = bilin + ((k & 1) ? ST0 : 0) + ((k & 2) ? ST1 : 0)
                          + ((k & 4) ? ST2 : 0) + ((k & 8) ? 1 : 0);
    const float wk = ((k & 1) ? w1[0] : w0[0]) * ((k & 2) ? w1[1] : w0[1])
                   * ((k & 4) ? w1[2] : w0[2]) * ((k & 8) ? w1[3] : w0[3]);
#pragma unroll
    for (int c = 0; c < NCH; ++c)
      acc[c] = fmaf(wk, g[(size_t)c * GP + lin], acc[c]);
  }
  float4* o = (float4*)(out + (size_t)n * NCH);
#pragma unroll
  for (int j = 0; j < 4; ++j) {
    float4 q;
    q.x = acc[4 * j + 0]; q.y = acc[4 * j + 1]; q.z = acc[4 * j + 2]; q.w = acc[4 * j + 3];
    o[j] = q;
  }
}

extern "C" void kernel_launch(void* const* d_in, const int* in_sizes, int n_in,
                              void* d_out, int out_size, void* d_ws, size_t ws_size,
                              hipStream_t stream) {
  const float* ray  = (const float*)d_in[0];
  const float* grid = (const float*)d_in[1];
  const float* rmin = (const float*)d_in[2];
  const float* rmax = (const float*)d_in[3];
  float* out = (float*)d_out;
  const int N = in_sizes[0] / 4;

  const size_t need = (size_t)GP * NCH * sizeof(float);   // 339.7 MB channels-last copy
  if (ws_size >= need && (N % 64) == 0) {
    float* gt = (float*)d_ws;
    lf4d_transpose<<<GP / 256, 256, 0, stream>>>(grid, gt);
    lf4d_wmma<<<N / 64, 64, 0, stream>>>(ray, gt, rmin, rmax, out);
  } else {
    lf4d_direct<<<(N + 255) / 256, 256, 0, stream>>>(ray, grid, rmin, rmax, out, N);
  }
}